// Model_rits_14714557956467
// MI455X (gfx1250) — compile-verified
//
#include <hip/hip_runtime.h>
#include <hip/hip_bf16.h>
#include <math.h>

// Problem constants (match the reference)
#define BB 1024
#define TT 256
#define NN 32
#define HH 64
#define EPSF 1e-5f

#define LDN 36   // padded row stride for 16xN tiles (bank-conflict relief)
#define LDH 68   // padded row stride for 16xH tiles

typedef float v2f __attribute__((ext_vector_type(2)));
typedef float v8f __attribute__((ext_vector_type(8)));

// ---------------------------------------------------------------------------
// WMMA helpers: D = A[16xK] * W^T(:,col0:col0+16) + acc  using
// V_WMMA_F32_16X16X4_F32 over K/4 chunks.
//  A: row-major, stride lda (LDS).   W: row-major [nout x K], stride ldw.
//  B-matrix element B[k][n] = W[col0+n][k].
// A-frag layout: lane m=lane&15, holds K = kchunk + 2*(lane>>4) + {0,1}
// B-frag layout: lane n=lane&15, holds same K pair.
// C/D  layout: lane n=lane&15, vgpr v holds row m = v + 8*(lane>>4).
// ---------------------------------------------------------------------------
__device__ __forceinline__ v8f gemm_tile(const float* __restrict__ A, int lda,
                                         const float* __restrict__ W, int ldw,
                                         int col0, int K, v8f acc) {
  const int lane = threadIdx.x & 31;
  const int m = lane & 15;
  const int kh = (lane >> 4) << 1;
  const float* Arow = A + m * lda + kh;
  const float* Wrow = W + (col0 + m) * ldw + kh;   // col == col0 + (lane&15)
  for (int k = 0; k < K; k += 4) {
    v2f a; a.x = Arow[k]; a.y = Arow[k + 1];
    v2f b; b.x = Wrow[k]; b.y = Wrow[k + 1];
    acc = __builtin_amdgcn_wmma_f32_16x16x4_f32(false, a, false, b,
                                                (short)0, acc, false, false);
  }
  return acc;
}

// Same, but zeroes the diagonal of W (FeatureRegression mask = 1 - I).
__device__ __forceinline__ v8f gemm_tile_offdiag(const float* __restrict__ A, int lda,
                                                 const float* __restrict__ W, int ldw,
                                                 int col0, int K, v8f acc) {
  const int lane = threadIdx.x & 31;
  const int m = lane & 15;
  const int kh = (lane >> 4) << 1;
  const int col = col0 + m;
  const float* Arow = A + m * lda + kh;
  for (int k = 0; k < K; k += 4) {
    const int k0 = k + kh;
    v2f a; a.x = Arow[k]; a.y = Arow[k + 1];
    v2f b;
    b.x = (k0     == col) ? 0.f : W[col * ldw + k0];
    b.y = (k0 + 1 == col) ? 0.f : W[col * ldw + k0 + 1];
    acc = __builtin_amdgcn_wmma_f32_16x16x4_f32(false, a, false, b,
                                                (short)0, acc, false, false);
  }
  return acc;
}

__device__ __forceinline__ void cfrag_store(float* dst, int ldd, int col0, v8f c) {
  const int lane = threadIdx.x & 31;
  const int n = lane & 15;
  const int mb = (lane >> 4) << 3;
#pragma unroll
  for (int v = 0; v < 8; ++v) dst[(mb + v) * ldd + col0 + n] = c[v];
}

__device__ __forceinline__ float sigmf(float x) { return 1.f / (1.f + expf(-x)); }

// ---------------------------------------------------------------------------
// Kernel 1: per-timestep mask-sum normalizer  ws_msum_inv[t] = 1/(sum(m)+eps)
// ---------------------------------------------------------------------------
__global__ __launch_bounds__(256) void rits_msum(const float* __restrict__ masks,
                                                 float* __restrict__ msum_inv) {
  const int t = blockIdx.x;
  __shared__ float red[256];
  float s = 0.f;
  for (int idx = threadIdx.x; idx < BB * NN; idx += 256) {
    int b = idx >> 5, n = idx & 31;
    s += masks[(size_t)b * TT * NN + (size_t)t * NN + n];
  }
  red[threadIdx.x] = s;
  __syncthreads();
  for (int st = 128; st > 0; st >>= 1) {
    if (threadIdx.x < st) red[threadIdx.x] += red[threadIdx.x + st];
    __syncthreads();
  }
  if (threadIdx.x == 0) msum_inv[t] = 1.0f / (red[0] + EPSF);
}

// ---------------------------------------------------------------------------
// Kernel 2: main recurrence. 64 blocks x 128 threads; block = 16 batch rows.
// ---------------------------------------------------------------------------
__global__ __launch_bounds__(128) void rits_main(
    const float* __restrict__ values, const float* __restrict__ masks,
    const float* __restrict__ deltas, const float* __restrict__ labels,
    const float* __restrict__ is_train,
    const float* __restrict__ td_h_W, const float* __restrict__ td_h_b,
    const float* __restrict__ td_x_W, const float* __restrict__ td_x_b,
    const float* __restrict__ hist_W, const float* __restrict__ hist_b,
    const float* __restrict__ feat_W, const float* __restrict__ feat_b,
    const float* __restrict__ wc_W, const float* __restrict__ wc_b,
    const float* __restrict__ W_ih, const float* __restrict__ W_hh,
    const float* __restrict__ b_ih, const float* __restrict__ b_hh,
    const float* __restrict__ out_W, const float* __restrict__ out_b,
    const float* __restrict__ msum_inv,
    float* __restrict__ out_pred, float* __restrict__ out_imp,
    float* __restrict__ ws_xloss, float* __restrict__ ws_yloss) {
  __shared__ float sh[16 * LDH];     // hidden state h
  __shared__ float sc[16 * LDH];     // cell state c
  __shared__ float sinp[16 * LDH];   // concat buffers: [gamma_x,m] then [c_c,m]
  __shared__ float sx[16 * LDN], smk[16 * LDN], sd[16 * LDN];
  __shared__ float sxh[16 * LDN], sxc[16 * LDN], szh[16 * LDN], sal[16 * LDN];
  __shared__ float red[128];

  const int tid = threadIdx.x;
  const int wv = tid >> 5;                    // wave id 0..3
  const int lane = tid & 31;
  const int fn = lane & 15;                   // fragment column
  const int fmb = (lane >> 4) << 3;           // fragment row base
  const int r0 = blockIdx.x * 16;             // batch-row base

  for (int i = tid; i < 16 * LDH; i += 128) { sh[i] = 0.f; sc[i] = 0.f; }
  __syncthreads();

  float xpart = 0.f;
  const v8f vz = {0.f, 0.f, 0.f, 0.f, 0.f, 0.f, 0.f, 0.f};

  for (int t = 0; t < TT; ++t) {
    const float inv = msum_inv[t];

    // ---- load x,m,d; gamma_x -> sinp[:, :32], m -> sinp[:, 32:64] ----
    for (int idx = tid; idx < 256 * 2; idx += 128) {
      int i = idx >> 5, n = idx & 31;
      size_t g = (size_t)(r0 + i) * TT * NN + (size_t)t * NN + n;
      float xv = values[g], mv = masks[g], dv = deltas[g];
      sx[i * LDN + n] = xv;
      smk[i * LDN + n] = mv;
      sd[i * LDN + n] = dv;
      float gx = expf(-fmaxf(dv * td_x_W[n * NN + n] + td_x_b[n], 0.f));
      sinp[i * LDH + n] = gx;
      sinp[i * LDH + 32 + n] = mv;
    }
    __syncthreads();

    // ---- Stage A/B: gamma_h tile wv;  h *= gamma_h  (fused) ----
    {
      v8f g = gemm_tile(sd, LDN, td_h_W, NN, wv * 16, NN, vz);
      float bias = td_h_b[wv * 16 + fn];
#pragma unroll
      for (int v = 0; v < 8; ++v) {
        float gh = expf(-fmaxf(g[v] + bias, 0.f));
        sh[(fmb + v) * LDH + wv * 16 + fn] *= gh;
      }
    }
    __syncthreads();

    // ---- Stage C: waves 0,1 -> x_h = h @ hist_W^T + b
    //              waves 2,3 -> alpha = [gamma_x,m] @ wc_W^T + b ----
    if (wv < 2) {
      v8f acc = gemm_tile(sh, LDH, hist_W, HH, wv * 16, HH, vz);
      float bias = hist_b[wv * 16 + fn];
#pragma unroll
      for (int v = 0; v < 8; ++v) acc[v] += bias;
      cfrag_store(sxh, LDN, wv * 16, acc);
    } else {
      int cw = wv - 2;
      v8f acc = gemm_tile(sinp, LDH, wc_W, 2 * NN, cw * 16, 2 * NN, vz);
      float bias = wc_b[cw * 16 + fn];
#pragma unroll
      for (int v = 0; v < 8; ++v) acc[v] += bias;
      cfrag_store(sal, LDN, cw * 16, acc);
    }
    __syncthreads();

    // ---- Stage D: x_c = m*x + (1-m)*x_h ; loss term 1 ----
    for (int idx = tid; idx < 512; idx += 128) {
      int i = idx >> 5, n = idx & 31;
      float xv = sx[i * LDN + n], mv = smk[i * LDN + n], xh = sxh[i * LDN + n];
      xpart += fabsf(xv - xh) * mv * inv;
      sxc[i * LDN + n] = mv * xv + (1.f - mv) * xh;
    }
    __syncthreads();

    // ---- Stage E: z_h = x_c @ (feat_W*(1-I))^T + b (waves 0,1) ----
    if (wv < 2) {
      v8f acc = gemm_tile_offdiag(sxc, LDN, feat_W, NN, wv * 16, NN, vz);
      float bias = feat_b[wv * 16 + fn];
#pragma unroll
      for (int v = 0; v < 8; ++v) acc[v] += bias;
      cfrag_store(szh, LDN, wv * 16, acc);
    }
    __syncthreads();

    // ---- Stage G: c_h, c_c, loss terms 2/3, imputation store, LSTM input ----
    for (int idx = tid; idx < 512; idx += 128) {
      int i = idx >> 5, n = idx & 31;
      float xv = sx[i * LDN + n], mv = smk[i * LDN + n];
      float xh = sxh[i * LDN + n], zh = szh[i * LDN + n], al = sal[i * LDN + n];
      xpart += fabsf(xv - zh) * mv * inv;
      float ch = al * zh + (1.f - al) * xh;
      xpart += fabsf(xv - ch) * mv * inv;
      float cc = mv * xv + (1.f - mv) * ch;
      sinp[i * LDH + n] = cc;
      sinp[i * LDH + 32 + n] = mv;
      out_imp[(size_t)(r0 + i) * TT * NN + (size_t)t * NN + n] = cc;
    }
    __syncthreads();

    // ---- Stage H: gates = [c_c,m]@W_ih^T + h@W_hh^T + b ; LSTM update ----
    v8f gate[4];
#pragma unroll
    for (int gx = 0; gx < 4; ++gx) {
      const int col0 = gx * 64 + wv * 16;
      v8f acc = gemm_tile(sinp, LDH, W_ih, 2 * NN, col0, 2 * NN, vz);
      acc = gemm_tile(sh, LDH, W_hh, HH, col0, HH, acc);
      float bias = b_ih[col0 + fn] + b_hh[col0 + fn];
#pragma unroll
      for (int v = 0; v < 8; ++v) acc[v] += bias;
      gate[gx] = acc;
    }
    __syncthreads();  // all waves done reading sh before overwriting it
#pragma unroll
    for (int v = 0; v < 8; ++v) {
      const int off = (fmb + v) * LDH + wv * 16 + fn;
      float cprev = sc[off];
      float iv = sigmf(gate[0][v]);
      float fv = sigmf(gate[1][v]);
      float gv = tanhf(gate[2][v]);
      float ov = sigmf(gate[3][v]);
      float cn = fv * cprev + iv * gv;
      sc[off] = cn;
      sh[off] = ov * tanhf(cn);
    }
    __syncthreads();
  }

  // ---- block-reduce x_loss partial ----
  red[tid] = xpart;
  __syncthreads();
  for (int s = 64; s > 0; s >>= 1) {
    if (tid < s) red[tid] += red[tid + s];
    __syncthreads();
  }
  if (tid == 0) ws_xloss[blockIdx.x] = red[0];
  __syncthreads();

  // ---- output head: logits, predictions, per-row BCE * is_train ----
  float yp = 0.f;
  if (tid < 16) {
    float acc = out_b[0];
    for (int k = 0; k < HH; ++k) acc += sh[tid * LDH + k] * out_W[k];
    out_pred[r0 + tid] = 1.f / (1.f + expf(-acc));
    float lab = labels[r0 + tid];
    float mx = fmaxf(-acc, 0.f);
    float bce = acc - acc * lab + mx + logf(expf(-mx) + expf(-acc - mx));
    yp = bce * is_train[r0 + tid];
  }
  red[tid] = yp;
  __syncthreads();
  for (int s = 64; s > 0; s >>= 1) {
    if (tid < s) red[tid] += red[tid + s];
    __syncthreads();
  }
  if (tid == 0) ws_yloss[blockIdx.x] = red[0];
}

// ---------------------------------------------------------------------------
// Kernel 3: deterministic final combine -> d_out[0] = loss
// ---------------------------------------------------------------------------
__global__ __launch_bounds__(64) void rits_final(const float* __restrict__ ws_xloss,
                                                 const float* __restrict__ ws_yloss,
                                                 const float* __restrict__ is_train,
                                                 float* __restrict__ out_loss) {
  if (threadIdx.x == 0 && blockIdx.x == 0) {
    float xl = 0.f, yl = 0.f, sit = 0.f;
    for (int i = 0; i < 64; ++i) { xl += ws_xloss[i]; yl += ws_yloss[i]; }
    for (int b = 0; b < BB; ++b) sit += is_train[b];
    out_loss[0] = xl / (float)TT + (yl / (sit + EPSF)) * 0.1f;
  }
}

// ---------------------------------------------------------------------------
extern "C" void kernel_launch(void* const* d_in, const int* in_sizes, int n_in,
                              void* d_out, int out_size, void* d_ws, size_t ws_size,
                              hipStream_t stream) {
  const float* values   = (const float*)d_in[0];
  const float* masks    = (const float*)d_in[1];
  const float* deltas   = (const float*)d_in[2];
  // d_in[3] evals, d_in[4] eval_masks: unused by reference outputs
  const float* labels   = (const float*)d_in[5];
  const float* is_train = (const float*)d_in[6];
  const float* td_h_W   = (const float*)d_in[7];
  const float* td_h_b   = (const float*)d_in[8];
  const float* td_x_W   = (const float*)d_in[9];
  const float* td_x_b   = (const float*)d_in[10];
  const float* hist_W   = (const float*)d_in[11];
  const float* hist_b   = (const float*)d_in[12];
  const float* feat_W   = (const float*)d_in[13];
  const float* feat_b   = (const float*)d_in[14];
  const float* wc_W     = (const float*)d_in[15];
  const float* wc_b     = (const float*)d_in[16];
  const float* W_ih     = (const float*)d_in[17];
  const float* W_hh     = (const float*)d_in[18];
  const float* b_ih     = (const float*)d_in[19];
  const float* b_hh     = (const float*)d_in[20];
  const float* out_W    = (const float*)d_in[21];
  const float* out_b    = (const float*)d_in[22];

  float* out = (float*)d_out;
  float* out_loss = out;            // [1]
  float* out_pred = out + 1;        // [B]
  float* out_imp  = out + 1 + BB;   // [B,T,N]

  float* ws = (float*)d_ws;
  float* msum_inv = ws;             // [T]
  float* ws_xloss = ws + TT;        // [64]
  float* ws_yloss = ws + TT + 64;   // [64]

  rits_msum<<<TT, 256, 0, stream>>>(masks, msum_inv);
  rits_main<<<BB / 16, 128, 0, stream>>>(
      values, masks, deltas, labels, is_train,
      td_h_W, td_h_b, td_x_W, td_x_b, hist_W, hist_b, feat_W, feat_b,
      wc_W, wc_b, W_ih, W_hh, b_ih, b_hh, out_W, out_b,
      msum_inv, out_pred, out_imp, ws_xloss, ws_yloss);
  rits_final<<<1, 64, 0, stream>>>(ws_xloss, ws_yloss, is_train, out_loss);
}